// ODEFunc_2465311228275
// MI455X (gfx1250) — compile-verified
//
#include <hip/hip_runtime.h>
#include <hip/hip_bf16.h>

// -------- types for the fp32 WMMA path (V_WMMA_F32_16X16X4_F32) ----------
typedef float v2f __attribute__((ext_vector_type(2)));
typedef float v8f __attribute__((ext_vector_type(8)));

#define ACT_NONE 0
#define ACT_TANH 1

__device__ __forceinline__ float fast_tanh(float x) {
    // tanh(x) = 1 - 2/(exp(2x)+1); exp via native v_exp_f32 (exp2)
    float e = __builtin_amdgcn_exp2f(x * 2.8853900817779268f); // 2*log2(e)
    return 1.0f - 2.0f * __builtin_amdgcn_rcpf(e + 1.0f);
}

// ---------------------------------------------------------------------------
// Generic D[M,N] = act(alpha * X[M,K] @ W[N,K]^T + bias[N])
// One 16x16 output tile per wave, K stepped by 4 via v_wmma_f32_16x16x4_f32.
// ISA layouts (CDNA5 ISA §7.12.2):
//   A 16x4 : lane l -> row l%16, VGPR v -> K = 2*(l/16)+v
//   B 4x16 : lane l -> col l%16, VGPR v -> K = 2*(l/16)+v   (B = W^T tile)
//   C/D    : lane l -> col l%16, VGPR v -> row v + 8*(l/16)
// Tile index is wave-uniform -> EXEC stays all-1s around WMMA.
// ---------------------------------------------------------------------------
template<int ACT>
__global__ void wmma_gemm_bias(const float* __restrict__ X, int ldx,
                               const float* __restrict__ W, int ldw,
                               const float* __restrict__ bias,
                               float* __restrict__ D, int ldd,
                               int M, int N, int K, float alpha)
{
    const int lane  = threadIdx.x & 31;
    const int wave  = threadIdx.x >> 5;
    const int waves = blockDim.x >> 5;
    const int tile  = blockIdx.x * waves + wave;
    const int tilesN = N >> 4;
    const int tm = tile / tilesN;
    const int tn = tile - tm * tilesN;
    const int m0 = tm << 4;
    const int n0 = tn << 4;
    if (m0 >= M) return;                       // wave-uniform exit

    const int half = lane >> 4;                // 0 or 1
    const int r    = lane & 15;
    const float* a_ptr = X + (size_t)(m0 + r) * ldx + 2 * half;
    const float* b_ptr = W + (size_t)(n0 + r) * ldw + 2 * half;

    v8f acc = {};
#pragma unroll 8
    for (int k = 0; k < K; k += 4) {
        v2f a = *(const v2f*)(a_ptr + k);
        v2f b = *(const v2f*)(b_ptr + k);
        acc = __builtin_amdgcn_wmma_f32_16x16x4_f32(
            /*neg_a=*/false, a, /*neg_b=*/false, b,
            /*c_mod=*/(short)0, acc, /*reuse_a=*/false, /*reuse_b=*/false);
    }

    const float bv = bias ? bias[n0 + r] : 0.0f;
#pragma unroll
    for (int v = 0; v < 8; ++v) {
        float val = acc[v] * alpha + bv;
        if (ACT == ACT_TANH) val = fast_tanh(val);
        D[(size_t)(m0 + v + 8 * half) * ldd + (n0 + r)] = val;
    }
}

// ---------------------------------------------------------------------------
// Row-wise masked softmax, in place. One block (256 threads) per (b,i) row of
// Nn=512 scores; diagonal j==i masked to -1e30 before max/exp/sum.
// ---------------------------------------------------------------------------
__global__ void softmax_mask_kernel(float* __restrict__ S, int Nn)
{
    const int bi = blockIdx.x;
    const int i  = bi % Nn;
    float* row = S + (size_t)bi * Nn;
    const int t = threadIdx.x;                 // 256 threads, 2 cols each
    __shared__ float red[16];

    const int j0 = t, j1 = t + 256;
    float x0 = (j0 == i) ? -1e30f : row[j0];
    float x1 = (j1 == i) ? -1e30f : row[j1];

    // block max
    float m = fmaxf(x0, x1);
    for (int off = 16; off; off >>= 1) m = fmaxf(m, __shfl_xor(m, off, 32));
    if ((t & 31) == 0) red[t >> 5] = m;
    __syncthreads();
    m = red[0];
#pragma unroll
    for (int w = 1; w < 8; ++w) m = fmaxf(m, red[w]);

    const float log2e = 1.4426950408889634f;
    float e0 = __builtin_amdgcn_exp2f((x0 - m) * log2e);
    float e1 = __builtin_amdgcn_exp2f((x1 - m) * log2e);

    // block sum
    float s = e0 + e1;
    for (int off = 16; off; off >>= 1) s += __shfl_xor(s, off, 32);
    __syncthreads();
    if ((t & 31) == 0) red[t >> 5] = s;
    __syncthreads();
    s = 0.0f;
#pragma unroll
    for (int w = 0; w < 8; ++w) s += red[w];

    const float inv = __builtin_amdgcn_rcpf(s);
    row[j0] = e0 * inv;
    row[j1] = e1 * inv;
}

// ---------------------------------------------------------------------------
// t[b,i,h] = sum_j attn[b,i,j] * tanh(zi[b,i,h] + zj[b,j,h] + b1[h])
// (W2/b2 hoisted out of the j-sum algebraically; applied later as a GEMM.)
// One block (128 threads == H) per (b,i); attn row staged through LDS in
// 128-wide chunks so the inner loop is: LDS broadcast + coalesced zj load +
// exp/rcp/fma. zj per batch = 256KB -> L2 resident across the whole sweep.
// ---------------------------------------------------------------------------
__global__ void rel_agg_kernel(const float* __restrict__ attn,
                               const float* __restrict__ zi,
                               const float* __restrict__ zj,
                               const float* __restrict__ b1,
                               float* __restrict__ t, int Nn, int H)
{
    const int bi = blockIdx.x;                 // 0 .. B*Nn-1
    const int b  = bi / Nn;
    const int h  = threadIdx.x;                // 0 .. H-1
    __shared__ float aw[128];

    const float base   = zi[(size_t)bi * H + h] + b1[h];
    const float* zjb   = zj + (size_t)b * Nn * H;
    const float* arow  = attn + (size_t)bi * Nn;

    float acc = 0.0f;
    for (int j0 = 0; j0 < Nn; j0 += 128) {
        __syncthreads();
        aw[h] = arow[j0 + h];
        __syncthreads();
#pragma unroll 4
        for (int jj = 0; jj < 128; ++jj) {
            const int j = j0 + jj;
            acc = fmaf(aw[jj], fast_tanh(base + zjb[(size_t)j * H + h]), acc);
        }
    }
    t[(size_t)bi * H + h] = acc;
}

// ---------------------------------------------------------------------------
extern "C" void kernel_launch(void* const* d_in, const int* in_sizes, int n_in,
                              void* d_out, int out_size, void* d_ws, size_t ws_size,
                              hipStream_t stream)
{
    const float* z   = (const float*)d_in[0];
    const float* s_t = (const float*)d_in[1];
    const float* W1  = (const float*)d_in[2];
    const float* b1  = (const float*)d_in[3];
    const float* W2  = (const float*)d_in[4];
    const float* b2  = (const float*)d_in[5];
    const float* Wq  = (const float*)d_in[6];
    const float* bq  = (const float*)d_in[7];
    const float* Wk  = (const float*)d_in[8];
    const float* bk  = (const float*)d_in[9];
    const float* W3  = (const float*)d_in[10];
    const float* b3  = (const float*)d_in[11];
    const float* W4  = (const float*)d_in[12];
    const float* b4  = (const float*)d_in[13];
    float* out = (float*)d_out;

    constexpr int Bb = 2, Nn = 512, H = 128, O = 32;
    const int M = Bb * Nn;                     // 1024 fused rows

    float* ws  = (float*)d_ws;
    float* q   = ws;                           // [M,H]
    float* k   = q   + (size_t)M * H;          // [M,H]
    float* zi  = k   + (size_t)M * H;          // [M,H]
    float* zjw = zi  + (size_t)M * H;          // [M,H]
    float* tt  = zjw + (size_t)M * H;          // [M,H]
    float* agg = tt  + (size_t)M * H;          // [M,H]
    float* h1  = agg + (size_t)M * H;          // [M,H]
    float* sc  = h1  + (size_t)M * H;          // [B,Nn,Nn]

    const dim3 blk(256);                       // 8 waves, 1 tile each
    const int gMH = (M / 16) * (H / 16) / 8;   // 64 blocks for 1024x128 outputs
    const int gNN = (Nn / 16) * (Nn / 16) / 8; // 128 blocks per 512x512 batch

    // q = z@Wq^T+bq ; k = s_t@Wk^T+bk ; zi = z@W1[:, :H]^T ; zj = z@W1[:, H:]^T
    wmma_gemm_bias<ACT_NONE><<<gMH, blk, 0, stream>>>(z,   H, Wq,     H,   bq,      q,   H, M, H, H, 1.0f);
    wmma_gemm_bias<ACT_NONE><<<gMH, blk, 0, stream>>>(s_t, O, Wk,     O,   bk,      k,   H, M, H, O, 1.0f);
    wmma_gemm_bias<ACT_NONE><<<gMH, blk, 0, stream>>>(z,   H, W1,     2*H, nullptr, zi,  H, M, H, H, 1.0f);
    wmma_gemm_bias<ACT_NONE><<<gMH, blk, 0, stream>>>(z,   H, W1 + H, 2*H, nullptr, zjw, H, M, H, H, 1.0f);

    // scores = (q @ k^T) / sqrt(H), per batch
    const float alpha = 0.08838834764831845f;  // 1/sqrt(128)
    for (int b = 0; b < Bb; ++b)
        wmma_gemm_bias<ACT_NONE><<<gNN, blk, 0, stream>>>(
            q + (size_t)b * Nn * H, H, k + (size_t)b * Nn * H, H, nullptr,
            sc + (size_t)b * Nn * Nn, Nn, Nn, Nn, H, alpha);

    // masked softmax -> attn (in place)
    softmax_mask_kernel<<<M, 256, 0, stream>>>(sc, Nn);

    // t = sum_j attn * tanh(zi + zj + b1)   (W2 hoisted out of the sum)
    rel_agg_kernel<<<M, 128, 0, stream>>>(sc, zi, zjw, b1, tt, Nn, H);

    // agg = t@W2^T+b2 ; h1 = tanh(agg@W3^T+b3) ; out = h1@W4^T+b4
    wmma_gemm_bias<ACT_NONE><<<gMH, blk, 0, stream>>>(tt,  H, W2, H, b2, agg, H, M, H, H, 1.0f);
    wmma_gemm_bias<ACT_TANH><<<gMH, blk, 0, stream>>>(agg, H, W3, H, b3, h1,  H, M, H, H, 1.0f);
    wmma_gemm_bias<ACT_NONE><<<gMH, blk, 0, stream>>>(h1,  H, W4, H, b4, out, H, M, H, H, 1.0f);
}